// NeighborhoodAttention1D_84911503442059
// MI455X (gfx1250) — compile-verified
//
#include <hip/hip_runtime.h>
#include <hip/hip_bf16.h>
#include <math.h>

typedef float v2f __attribute__((ext_vector_type(2)));
typedef float v8f __attribute__((ext_vector_type(8)));

#define EMBED_DIM 512
#define NUM_HEADS 8
#define HEAD_DIM 64
#define KERNEL_SZ 33
#define HALO 16
#define SEQ_LEN 4096
#define BATCH 4
#define ATT_SCALE 0.125f   /* 64^-0.5 */

/* ---------------- GEMM: C = A(MxK) * W(KxN) + bias ----------------
 * BM=128, BN=64, KT=64.  256 threads = 8 waves; wave w computes rows
 * m0+16w..+15 across all 64 columns (4 accumulators).  f32 WMMA 16x16x4.
 *
 * W panel lives in LDS in K-pair layout: word[(k>>1)*LDSROW + 2*n + (k&1)]
 * so each B fragment is one aligned ds_load_b64.  LDSROW=160 => the two
 * lane-halves (kpair 2s vs 2s+1) map to disjoint bank halves (160%64==32).
 * LDS W panel is double-buffered across K chunks; A fragments and the next
 * W chunk are prefetched into registers so WMMAs overlap memory latency.
 */
#define BM 128
#define BN 64
#define KT 64
#define LDSROW 160                 /* words per k-pair row (2*BN + 32 pad) */
#define CHUNKW ((KT / 2) * LDSROW) /* words per LDS buffer = 5120 (20 KB) */

__global__ __launch_bounds__(256)
void gemm_bias_f32(const float* __restrict__ A, const float* __restrict__ W,
                   const float* __restrict__ bias, float* __restrict__ C,
                   int M, int N, int K) {
  __shared__ float lw[2 * CHUNKW];
  const int tid  = threadIdx.x;
  const int wave = tid >> 5;
  const int lane = tid & 31;
  const int l15  = lane & 15;
  const int half = lane >> 4;

  const int n0 = blockIdx.x * BN;
  const int m0 = blockIdx.y * BM;
  const int mrow = m0 + wave * 16 + l15;

  /* staging coordinates: 256 threads cover 16 k-rows x 16 float4 per pass */
  const int sr  = tid >> 4;          /* 0..15 */
  const int sc4 = (tid & 15) * 4;

  v8f acc0 = {}, acc1 = {}, acc2 = {}, acc3 = {};
  float4 wreg[4];

  /* prefetch first W chunk into registers */
  #pragma unroll
  for (int p = 0; p < 4; ++p)
    wreg[p] = *(const float4*)(W + (size_t)(sr + p * 16) * N + n0 + sc4);

  const int nc = K / KT;
  int cur = 0;
  for (int c = 0; c < nc; ++c) {
    float* buf = lw + cur * CHUNKW;
    /* commit staged W chunk to LDS (k-pair layout) */
    #pragma unroll
    for (int p = 0; p < 4; ++p) {
      const int k = sr + p * 16;
      float* d = buf + (k >> 1) * LDSROW + (k & 1);
      d[2 * (sc4 + 0)] = wreg[p].x;
      d[2 * (sc4 + 1)] = wreg[p].y;
      d[2 * (sc4 + 2)] = wreg[p].z;
      d[2 * (sc4 + 3)] = wreg[p].w;
    }
    __syncthreads();

    /* prefetch next W chunk while computing this one */
    if (c + 1 < nc) {
      const int k0n = (c + 1) * KT;
      #pragma unroll
      for (int p = 0; p < 4; ++p)
        wreg[p] = *(const float4*)(W + (size_t)(k0n + sr + p * 16) * N + n0 + sc4);
    }

    /* preload all A fragments for this chunk (16 independent b64 loads) */
    const float* Ap = A + (size_t)mrow * K + c * KT + half * 2;
    v2f af[16];
    #pragma unroll
    for (int i = 0; i < 16; ++i)
      af[i] = *(const v2f*)(Ap + 4 * i);

    /* inner wmma loop; B fragments double-buffered from LDS */
    v2f bb[2][4];
    const float* bbase = buf + 2 * l15;
    {
      const float* bp = bbase + half * LDSROW;
      bb[0][0] = *(const v2f*)(bp);
      bb[0][1] = *(const v2f*)(bp + 32);
      bb[0][2] = *(const v2f*)(bp + 64);
      bb[0][3] = *(const v2f*)(bp + 96);
    }
    #pragma unroll
    for (int s = 0; s < 16; ++s) {
      if (s < 15) {
        const float* bp = bbase + ((s + 1) * 2 + half) * LDSROW;
        bb[(s + 1) & 1][0] = *(const v2f*)(bp);
        bb[(s + 1) & 1][1] = *(const v2f*)(bp + 32);
        bb[(s + 1) & 1][2] = *(const v2f*)(bp + 64);
        bb[(s + 1) & 1][3] = *(const v2f*)(bp + 96);
      }
      acc0 = __builtin_amdgcn_wmma_f32_16x16x4_f32(false, af[s], false, bb[s & 1][0], (short)0, acc0, false, false);
      acc1 = __builtin_amdgcn_wmma_f32_16x16x4_f32(false, af[s], false, bb[s & 1][1], (short)0, acc1, false, false);
      acc2 = __builtin_amdgcn_wmma_f32_16x16x4_f32(false, af[s], false, bb[s & 1][2], (short)0, acc2, false, false);
      acc3 = __builtin_amdgcn_wmma_f32_16x16x4_f32(false, af[s], false, bb[s & 1][3], (short)0, acc3, false, false);
    }
    cur ^= 1;
  }

  /* D layout: VGPR r -> row r (lanes 0-15) / r+8 (lanes 16-31); N = lane%16 */
  const int crow = m0 + wave * 16 + half * 8;
  const float bb0 = bias[n0 + l15];
  const float bb1 = bias[n0 + 16 + l15];
  const float bb2 = bias[n0 + 32 + l15];
  const float bb3 = bias[n0 + 48 + l15];
  #pragma unroll
  for (int r = 0; r < 8; ++r) {
    float* outp = C + (size_t)(crow + r) * N + n0 + l15;
    outp[0]  = acc0[r] + bb0;
    outp[16] = acc1[r] + bb1;
    outp[32] = acc2[r] + bb2;
    outp[48] = acc3[r] + bb3;
  }
}

/* ---------------- Neighborhood attention, one query per thread ----------------
 * 64 queries per block for one (b,h); union of windows spans <= 96 keys,
 * staged in LDS with stride 65 (bank = (row+d)%64 -> conflict-free).
 */
#define TL 64
#define MAXSPAN 96
#define LDA 65

__device__ __forceinline__ int win_start(int l) {
  int s = l - HALO;
  s = s < 0 ? 0 : s;
  const int hi = SEQ_LEN - KERNEL_SZ;
  return s > hi ? hi : s;
}

__global__ __launch_bounds__(TL)
void natt1d(const float* __restrict__ qkv, float* __restrict__ out) {
  __shared__ float ks[MAXSPAN * LDA];
  __shared__ float vs[MAXSPAN * LDA];
  const int tid = threadIdx.x;
  const int t0  = blockIdx.x * TL;
  const int h   = blockIdx.y;
  const int b   = blockIdx.z;
  const size_t rowstride = 3 * EMBED_DIM;       /* 1536 */
  const size_t basebl = (size_t)b * SEQ_LEN;

  const int sp0 = win_start(t0);
  const int cnt = win_start(t0 + TL - 1) + KERNEL_SZ - sp0;   /* <= 96 */

  /* stage K and V spans */
  for (int idx = tid; idx < cnt * (HEAD_DIM / 4); idx += TL) {
    const int row = idx >> 4;
    const int c4  = (idx & 15) * 4;
    const size_t g = (basebl + sp0 + row) * rowstride + (size_t)h * HEAD_DIM + c4;
    const float4 kk4 = *(const float4*)(qkv + EMBED_DIM + g);
    const float4 vv4 = *(const float4*)(qkv + 2 * EMBED_DIM + g);
    ks[row * LDA + c4 + 0] = kk4.x; ks[row * LDA + c4 + 1] = kk4.y;
    ks[row * LDA + c4 + 2] = kk4.z; ks[row * LDA + c4 + 3] = kk4.w;
    vs[row * LDA + c4 + 0] = vv4.x; vs[row * LDA + c4 + 1] = vv4.y;
    vs[row * LDA + c4 + 2] = vv4.z; vs[row * LDA + c4 + 3] = vv4.w;
  }
  __syncthreads();

  const int l = t0 + tid;
  const size_t qoff = (basebl + l) * rowstride + (size_t)h * HEAD_DIM;
  float q[HEAD_DIM];
  #pragma unroll
  for (int d = 0; d < HEAD_DIM; d += 4) {
    const float4 t = *(const float4*)(qkv + qoff + d);
    q[d] = t.x; q[d + 1] = t.y; q[d + 2] = t.z; q[d + 3] = t.w;
  }

  const int js = win_start(l) - sp0;
  float logits[KERNEL_SZ];
  float mx = -INFINITY;
  #pragma unroll
  for (int j = 0; j < KERNEL_SZ; ++j) {
    const float* kp = &ks[(js + j) * LDA];
    float dot = 0.f;
    #pragma unroll
    for (int d = 0; d < HEAD_DIM; ++d) dot += q[d] * kp[d];
    dot *= ATT_SCALE;
    logits[j] = dot;
    mx = fmaxf(mx, dot);
  }
  float s = 0.f;
  #pragma unroll
  for (int j = 0; j < KERNEL_SZ; ++j) {
    const float p = __expf(logits[j] - mx);
    logits[j] = p;
    s += p;
  }
  const float inv = 1.f / s;

  const size_t ooff = (basebl + l) * (size_t)EMBED_DIM + (size_t)h * HEAD_DIM;
  #pragma unroll 4
  for (int d = 0; d < HEAD_DIM; ++d) {
    float accv = 0.f;
    #pragma unroll
    for (int j = 0; j < KERNEL_SZ; ++j) accv += logits[j] * vs[(js + j) * LDA + d];
    out[ooff + d] = accv * inv;
  }
}

extern "C" void kernel_launch(void* const* d_in, const int* in_sizes, int n_in,
                              void* d_out, int out_size, void* d_ws, size_t ws_size,
                              hipStream_t stream) {
  const float* x      = (const float*)d_in[0];
  const float* w_qkv  = (const float*)d_in[1];
  const float* b_qkv  = (const float*)d_in[2];
  const float* w_proj = (const float*)d_in[3];
  const float* b_proj = (const float*)d_in[4];
  float* out = (float*)d_out;

  const int M = BATCH * SEQ_LEN;                  /* 16384 */
  float* qkv  = (float*)d_ws;                     /* 16384 x 1536 */
  float* attn = qkv + (size_t)M * 3 * EMBED_DIM;  /* 16384 x 512  */

  gemm_bias_f32<<<dim3((3 * EMBED_DIM) / BN, M / BM), 256, 0, stream>>>(
      x, w_qkv, b_qkv, qkv, M, 3 * EMBED_DIM, EMBED_DIM);

  natt1d<<<dim3(SEQ_LEN / TL, NUM_HEADS, BATCH), TL, 0, stream>>>(qkv, attn);

  gemm_bias_f32<<<dim3(EMBED_DIM / BN, M / BM), 256, 0, stream>>>(
      attn, w_proj, b_proj, out, M, EMBED_DIM, EMBED_DIM);
}